// TokenEmbedding_55997783605696
// MI455X (gfx1250) — compile-verified
//
#include <hip/hip_runtime.h>

// Dilated causal conv (m=7, tau=3, 74 kernels of 8x3) as WMMA f32 16x16x4 GEMM.
// out[b,t,c*73+k] = sum_{d=0..23} W2[d][k] * x[b, t + d - 22, c]  (+ edge zeroing)

constexpr int kB   = 32;
constexpr int kL   = 4096;
constexpr int kC   = 7;
constexpr int kNK  = 74;   // real kernels
constexpr int kKD  = 24;   // reduction taps (offsets -22..+1)
constexpr int kNT  = 80;   // kernels padded to 5 tiles of 16
constexpr int kTT  = 128;  // timesteps per block
constexpr int kXSL = 151;  // staged x per channel: [-22 .. +128]
constexpr int kXSS = 152;  // padded stride
constexpr int kOC  = 512;  // output channels

typedef __attribute__((ext_vector_type(2))) float v2f;
typedef __attribute__((ext_vector_type(8))) float v8f;

__launch_bounds__(256)
__global__ void dilated_conv_wmma(const float* __restrict__ x,
                                  const float* __restrict__ kern,
                                  float* __restrict__ out) {
  __shared__ float W2[kKD][kNT];        // 24 x 80 re-indexed filter bank
  __shared__ float xs[kC][kXSS];        // staged x with halo, zero-padded

  const int tid = threadIdx.x;
  const int bx  = blockIdx.x;
  const int b   = bx / (kL / kTT);
  const int T0  = (bx % (kL / kTT)) * kTT;

  // ---- Stage W2: W2[d][k] = kern[k, i, j] with offset o = d-22 = (j-1) - 3i
  for (int e = tid; e < kKD * kNT; e += 256) {
    int k = e % kNT;
    int d = e / kNT;
    int o = d - 22;
    int r = ((o % 3) + 3) % 3;          // o mod 3 in [0,3)
    int jm1 = (r == 2) ? -1 : r;        // j-1 in {-1,0,1}
    int j = jm1 + 1;
    int i = (jm1 - o) / 3;              // in [0,8)
    float v = 0.0f;
    if (k < kNK) v = kern[(k * 8 + i) * 3 + j];
    W2[d][k] = v;
  }

  // ---- Stage x tile (+22 halo in front, +1 beyond), zero-padded
  for (int e = tid; e < kC * kXSL; e += 256) {
    int c = e / kXSL;
    int i = e % kXSL;
    int g = T0 - 22 + i;
    float v = 0.0f;
    if (g >= 0 && g < kL) v = x[((size_t)b * kL + g) * kC + c];
    xs[c][i] = v;
  }
  __syncthreads();

  const int wave = tid >> 5;
  const int lane = tid & 31;
  const int half = lane >> 4;           // A/B K-half, D row-half
  const int m    = lane & 15;           // A row / D column

  // ---- Hoist all B fragments (shared by every strip of this wave)
  // B 4x16 f32 layout: v0 = K(kk*4 + 2*half), v1 = K(kk*4 + 2*half + 1), col = lane&15
  v2f Bf[6][5];
  #pragma unroll
  for (int kk = 0; kk < 6; ++kk) {
    int d0 = kk * 4 + half * 2;
    #pragma unroll
    for (int nt = 0; nt < 5; ++nt) {
      Bf[kk][nt][0] = W2[d0][nt * 16 + m];
      Bf[kk][nt][1] = W2[d0 + 1][nt * 16 + m];
    }
  }

  // ---- 56 strips: (t_tile 0..7) x (channel 0..6); 7 per wave
  for (int s = wave; s < 56; s += 8) {
    int t_tile = s / 7;
    int c      = s % 7;
    int tbase  = t_tile * 16;
    int t_g    = T0 + tbase + m;        // global time of this A row

    // A 16x4 f32 layout: lanes 0-15 hold (K0,K1), lanes 16-31 hold (K2,K3), row = lane&15
    v2f Af[6];
    #pragma unroll
    for (int kk = 0; kk < 6; ++kk) {
      int d0 = kk * 4 + half * 2;
      float a0 = xs[c][tbase + m + d0];
      float a1 = xs[c][tbase + m + d0 + 1];
      // window-pad rule: zero tap when t + (j-1) >= L ; (d+2)%3 -> {0,1,-1}
      int r0 = (d0 + 2) % 3; int jm10 = (r0 == 2) ? -1 : r0;
      int r1 = (d0 + 3) % 3; int jm11 = (r1 == 2) ? -1 : r1;
      if (t_g + jm10 >= kL) a0 = 0.0f;
      if (t_g + jm11 >= kL) a1 = 0.0f;
      Af[kk][0] = a0;
      Af[kk][1] = a1;
    }

    #pragma unroll
    for (int nt = 0; nt < 5; ++nt) {
      v8f acc = {0.f, 0.f, 0.f, 0.f, 0.f, 0.f, 0.f, 0.f};
      #pragma unroll
      for (int kk = 0; kk < 6; ++kk) {
        acc = __builtin_amdgcn_wmma_f32_16x16x4_f32(
            false, Af[kk], false, Bf[kk][nt], (short)0, acc, false, false);
      }
      // D layout: col N = lane&15, rows M = vgpr + 8*half
      int k = nt * 16 + m;
      bool write = (k < 73) | ((k == 73) & (c == 0));
      if (write) {
        int oc = (k < 73) ? (c * 73 + k) : 511;   // last filter -> channel 0 slot
        int mb = half * 8;
        size_t base = ((size_t)b * kL + (T0 + tbase + mb)) * kOC + oc;
        #pragma unroll
        for (int r = 0; r < 8; ++r) {
          out[base + (size_t)r * kOC] = acc[r];
        }
      }
    }
  }
}

extern "C" void kernel_launch(void* const* d_in, const int* in_sizes, int n_in,
                              void* d_out, int out_size, void* d_ws, size_t ws_size,
                              hipStream_t stream) {
  const float* x    = (const float*)d_in[0];   // (32, 4096, 7) f32
  const float* kern = (const float*)d_in[1];   // (74, 8, 3)   f32
  float* out        = (float*)d_out;           // (32, 4096, 512) f32

  dim3 grid(kB * (kL / kTT));                  // 1024 blocks
  dim3 block(256);                             // 8 wave32
  dilated_conv_wmma<<<grid, block, 0, stream>>>(x, kern, out);
}